// SAGPool_32014686224552
// MI455X (gfx1250) — compile-verified
//
#include <hip/hip_runtime.h>
#include <math.h>

#define NPG     2048
#define KKEEP   1024
#define NNODES  131072
#define NEDGES  2097152
#define DIM     256
#define NGRAPH  64
#define NK      (NGRAPH * KKEEP)   // 65536 kept nodes total

typedef float v2f __attribute__((ext_vector_type(2)));
typedef float v8f __attribute__((ext_vector_type(8)));

// ---------------------------------------------------------------------------
// Init per-call scratch state: deg=1 (self loop), t=0, newid=-1
// ---------------------------------------------------------------------------
__global__ __launch_bounds__(256) void k_init(float* __restrict__ deg,
                                              float* __restrict__ t,
                                              int* __restrict__ newid) {
    int n = blockIdx.x * 256 + threadIdx.x;
    if (n < NNODES) {
        deg[n]   = 1.0f;   // self-loop contributes 1 to every node's degree
        t[n]     = 0.0f;
        newid[n] = -1;
    }
}

// ---------------------------------------------------------------------------
// h = X @ W  via V_WMMA_F32_16X16X4_F32  (one wave per 16-row tile)
// A-frag layout (16x4 f32): lanes 0-15 -> K=0(,1); lanes 16-31 -> K=2(,3); M=lane&15
// B-frag (4x16): column 0 carries W[k..k+3]; other columns zero.
// D column 0: lane 0 holds M=0..7 in acc[0..7], lane 16 holds M=8..15.
// ---------------------------------------------------------------------------
__global__ __launch_bounds__(256) void k_matvec(const float* __restrict__ X,
                                                const float* __restrict__ W,
                                                float* __restrict__ h) {
    __shared__ float sW[DIM];
    sW[threadIdx.x] = W[threadIdx.x];
    __syncthreads();

    const int wave = threadIdx.x >> 5;
    const int lane = threadIdx.x & 31;
    const int tile = blockIdx.x * 8 + wave;       // 8 waves per block
    const int rowb = tile * 16;
    const int m    = lane & 15;
    const int half = lane >> 4;                   // 0: K+{0,1}, 1: K+{2,3}

    const float* xp = X + (size_t)(rowb + m) * DIM + half * 2;

    v8f acc = {};
    #pragma unroll 4
    for (int k = 0; k < DIM; k += 4) {
        v2f a;
        a.x = xp[k];
        a.y = xp[k + 1];
        v2f bf;
        float w0 = sW[k + half * 2];
        float w1 = sW[k + half * 2 + 1];
        bf.x = (m == 0) ? w0 : 0.0f;
        bf.y = (m == 0) ? w1 : 0.0f;
        acc = __builtin_amdgcn_wmma_f32_16x16x4_f32(
            /*neg_a=*/false, a, /*neg_b=*/false, bf,
            /*c_mod=*/(short)0, acc, /*reuse_a=*/false, /*reuse_b=*/false);
    }

    if (m == 0) {
        int base = rowb + half * 8;
        #pragma unroll
        for (int v = 0; v < 8; ++v) h[base + v] = acc[v];
    }
}

// ---------------------------------------------------------------------------
// Degree accumulation over edge destinations (self-loop already in init)
// ---------------------------------------------------------------------------
__global__ __launch_bounds__(256) void k_deg(const int* __restrict__ dst,
                                             float* __restrict__ deg) {
    int e = blockIdx.x * 256 + threadIdx.x;
    if (e < NEDGES) atomicAdd(&deg[dst[e]], 1.0f);
}

// deg -> dinv in place (deg >= 1 always, so no zero-guard branch needed)
__global__ __launch_bounds__(256) void k_dinv(float* __restrict__ deg) {
    int n = blockIdx.x * 256 + threadIdx.x;
    if (n < NNODES) deg[n] = rsqrtf(deg[n]);
}

// t[d] += h[s] * dinv[s] for each edge
__global__ __launch_bounds__(256) void k_scatter(const int* __restrict__ src,
                                                 const int* __restrict__ dst,
                                                 const float* __restrict__ h,
                                                 const float* __restrict__ dinv,
                                                 float* __restrict__ t) {
    int e = blockIdx.x * 256 + threadIdx.x;
    if (e < NEDGES) {
        int s = src[e];
        atomicAdd(&t[dst[e]], h[s] * dinv[s]);
    }
}

// attn[n] = dinv[n]*(t[n] + h[n]*dinv[n]) + b   (in place into t)
__global__ __launch_bounds__(256) void k_finalize(const float* __restrict__ h,
                                                  const float* __restrict__ dinv,
                                                  float* __restrict__ t,
                                                  const float* __restrict__ b) {
    int n = blockIdx.x * 256 + threadIdx.x;
    if (n < NNODES) {
        float di = dinv[n];
        t[n] = di * (t[n] + h[n] * di) + b[0];
    }
}

// ---------------------------------------------------------------------------
// Per-graph exact top-K in descending-score order (matches jax.lax.top_k,
// ties -> smaller index first). 64-bit keys: [sortable_float | NPG-1-idx],
// bitonic-sorted descending in LDS. One workgroup (256 threads) per graph.
// ---------------------------------------------------------------------------
__global__ __launch_bounds__(256) void k_topk(const float* __restrict__ attn,
                                              const int* __restrict__ gi_in,
                                              int* __restrict__ perm,
                                              float* __restrict__ gate,
                                              int* __restrict__ newid,
                                              float* __restrict__ gi_out) {
    __shared__ unsigned long long keys[NPG];
    const int g   = blockIdx.x;
    const int tid = threadIdx.x;

    for (int i = tid; i < NPG; i += 256) {
        float s = attn[g * NPG + i];
        unsigned int u = __float_as_uint(s);
        u = (u & 0x80000000u) ? ~u : (u | 0x80000000u);   // monotone float->uint
        keys[i] = ((unsigned long long)u << 32) |
                  (unsigned long long)(unsigned int)(NPG - 1 - i);
    }
    __syncthreads();

    for (int k = 2; k <= NPG; k <<= 1) {
        for (int j = k >> 1; j > 0; j >>= 1) {
            for (int i = tid; i < NPG; i += 256) {
                int ixj = i ^ j;
                if (ixj > i) {
                    unsigned long long a = keys[i];
                    unsigned long long c = keys[ixj];
                    bool desc = ((i & k) == 0);
                    if (desc ? (a < c) : (a > c)) {
                        keys[i]   = c;
                        keys[ixj] = a;
                    }
                }
            }
            __syncthreads();
        }
    }

    for (int r = tid; r < KKEEP; r += 256) {
        unsigned long long key = keys[r];
        int li    = NPG - 1 - (int)(unsigned int)(key & 0xFFFFFFFFu);
        int node  = g * NPG + li;
        int out_r = g * KKEEP + r;
        perm[out_r]  = node;
        newid[node]  = out_r;
        gate[out_r]  = tanhf(attn[node]);
        gi_out[out_r] = (float)gi_in[node];
    }
}

// ---------------------------------------------------------------------------
// new_x[r,:] = X[perm[r],:] * gate[r]   (float4 vectorized, 4 rows / block)
// ---------------------------------------------------------------------------
__global__ __launch_bounds__(256) void k_gather(const float* __restrict__ X,
                                                const int* __restrict__ perm,
                                                const float* __restrict__ gate,
                                                float* __restrict__ newx) {
    int idx = blockIdx.x * 256 + threadIdx.x;   // over NK*64 float4 slots
    int row = idx >> 6;
    int c   = idx & 63;
    float gv = gate[row];
    const float4* xs = (const float4*)X;
    float4 v = xs[(size_t)perm[row] * 64 + c];
    float4* o = (float4*)newx;
    o[(size_t)row * 64 + c] = make_float4(v.x * gv, v.y * gv, v.z * gv, v.w * gv);
}

// ---------------------------------------------------------------------------
// Edge remap: both endpoints kept -> new ids, else -1. Output as float.
// ---------------------------------------------------------------------------
__global__ __launch_bounds__(256) void k_edges(const int* __restrict__ src,
                                               const int* __restrict__ dst,
                                               const int* __restrict__ newid,
                                               float* __restrict__ out_src,
                                               float* __restrict__ out_dst) {
    int e = blockIdx.x * 256 + threadIdx.x;
    if (e < NEDGES) {
        int ns = newid[src[e]];
        int nd = newid[dst[e]];
        bool valid = (ns >= 0) && (nd >= 0);
        out_src[e] = valid ? (float)ns : -1.0f;
        out_dst[e] = valid ? (float)nd : -1.0f;
    }
}

// ---------------------------------------------------------------------------
extern "C" void kernel_launch(void* const* d_in, const int* in_sizes, int n_in,
                              void* d_out, int out_size, void* d_ws, size_t ws_size,
                              hipStream_t stream) {
    const float* X  = (const float*)d_in[0];
    const int*   gi = (const int*)d_in[1];
    const int*   ei = (const int*)d_in[2];
    const float* W  = (const float*)d_in[3];
    const float* b  = (const float*)d_in[4];
    const int* src = ei;
    const int* dst = ei + NEDGES;

    float* ws    = (float*)d_ws;
    float* h     = ws;                         // [N]
    float* deg   = ws + NNODES;                // [N] -> dinv (in place)
    float* t     = ws + 2 * (size_t)NNODES;    // [N] -> attn (in place)
    int*   newid = (int*)(ws + 3 * (size_t)NNODES);   // [N]
    int*   perm  = (int*)(ws + 4 * (size_t)NNODES);   // [NK]
    float* gate  = ws + 4 * (size_t)NNODES + NK;      // [NK]

    float* out_x   = (float*)d_out;                    // [NK*DIM]
    float* out_src = out_x + (size_t)NK * DIM;         // [E]
    float* out_dst = out_src + NEDGES;                 // [E]
    float* out_gi  = out_dst + NEDGES;                 // [NK]

    k_init    <<<NNODES / 256, 256, 0, stream>>>(deg, t, newid);
    k_matvec  <<<NNODES / 128, 256, 0, stream>>>(X, W, h);     // 8 waves * 16 rows
    k_deg     <<<NEDGES / 256, 256, 0, stream>>>(dst, deg);
    k_dinv    <<<NNODES / 256, 256, 0, stream>>>(deg);
    k_scatter <<<NEDGES / 256, 256, 0, stream>>>(src, dst, h, deg, t);
    k_finalize<<<NNODES / 256, 256, 0, stream>>>(h, deg, t, b);
    k_topk    <<<NGRAPH, 256, 0, stream>>>(t, gi, perm, gate, newid, out_gi);
    k_gather  <<<(NK * 64) / 256, 256, 0, stream>>>(X, perm, gate, out_x);
    k_edges   <<<NEDGES / 256, 256, 0, stream>>>(src, dst, newid, out_src, out_dst);
}